// GraphConv_20615843020934
// MI455X (gfx1250) — compile-verified
//
#include <hip/hip_runtime.h>
#include <hip/hip_bf16.h>

// GraphConv (norm='both' in-degree path, no weight):
//   h = scatter_add(feat[src] -> dst);  h *= rsqrt(max(in_deg, 1))
// N_NODES=10000, N_EDGES=640000, D_FEAT=128 (f32).
//
// Strategy: L2-resident gather + HW f32 atomics. Edge indices are staged to
// LDS with CDNA5 async global->LDS loads; feat rows read as b128 per lane;
// upcoming source rows prefetched with global_prefetch_b8.
//
// NOTE: gconv_scatter is defined FIRST so the disasm snippet shows it.

#define D_FEAT 128
#define EPB 128            // edges staged per block (256 threads, 8 waves)
#define EDGES_PER_WAVE (EPB / 8)

__device__ __forceinline__ void stage_b32_async(const int* gp, int* lp) {
#if __has_builtin(__builtin_amdgcn_global_load_async_to_lds_b32)
  // Per-lane async copy: 4 bytes global -> LDS, tracked by ASYNCcnt.
  __builtin_amdgcn_global_load_async_to_lds_b32((int*)gp, lp,
                                                /*offset=*/0, /*cpol=*/0);
#else
  *lp = *gp;  // fallback: plain LDS store path
#endif
}

__device__ __forceinline__ void stage_wait() {
#if __has_builtin(__builtin_amdgcn_global_load_async_to_lds_b32) && \
    __has_builtin(__builtin_amdgcn_s_wait_asynccnt)
  __builtin_amdgcn_s_wait_asynccnt(0);
#endif
}

__global__ void __launch_bounds__(256)
gconv_scatter(const float* __restrict__ feat,
              const int* __restrict__ src,
              const int* __restrict__ dst,
              float* __restrict__ h,
              float* __restrict__ deg,
              int n_edges) {
  __shared__ int s_idx[2 * EPB];  // [0,EPB): src indices, [EPB,2*EPB): dst

  const int tid  = threadIdx.x;        // 0..255
  const int base = blockIdx.x * EPB;

  // ---- stage this block's edge indices into LDS (async DMA path) ----
  {
    const int  which = tid >> 7;                 // 0 -> src, 1 -> dst
    const int  e     = tid & (EPB - 1);
    int ge = base + e;
    if (ge >= n_edges) ge = n_edges - 1;         // clamp; never consumed
    const int* gsrc = which ? dst : src;
    stage_b32_async(gsrc + ge, &s_idx[tid]);
  }
  stage_wait();        // my wave's ASYNCcnt == 0
  __syncthreads();     // all waves' staged indices visible

  const int wave = tid >> 5;    // 0..7
  const int lane = tid & 31;    // 0..31 (wave32)
  const int col  = lane * 4;    // each lane owns a 16B chunk of the 512B row

  // ---- per-wave edge loop: gather b128, 4x HW f32 atomics ----
  #pragma unroll 4
  for (int i = 0; i < EDGES_PER_WAVE; ++i) {
    const int e_loc = wave * EDGES_PER_WAVE + i;
    const int ge    = base + e_loc;
    if (ge >= n_edges) break;                    // uniform within the wave

    const int s = s_idx[e_loc];
    const int d = s_idx[EPB + e_loc];

    // Hide L2 gather latency: prefetch the source row two edges ahead.
    if (i + 2 < EDGES_PER_WAVE && base + e_loc + 2 < n_edges) {
      const int sp = s_idx[e_loc + 2];
      __builtin_prefetch(feat + (size_t)sp * D_FEAT + col, 0, 0);
    }

    const float4 v =
        *(const float4*)(feat + (size_t)s * D_FEAT + col);  // global_load_b128
    float* out = h + (size_t)d * D_FEAT + col;
    unsafeAtomicAdd(out + 0, v.x);   // global_atomic_add_f32
    unsafeAtomicAdd(out + 1, v.y);
    unsafeAtomicAdd(out + 2, v.z);
    unsafeAtomicAdd(out + 3, v.w);

    if (lane == 0) unsafeAtomicAdd(deg + d, 1.0f);
  }
}

__global__ void gconv_zero(float* __restrict__ h, float* __restrict__ deg,
                           int n_h, int n_deg) {
  int i = blockIdx.x * blockDim.x + threadIdx.x;
  if (i < n_h)   h[i]   = 0.0f;
  if (i < n_deg) deg[i] = 0.0f;
}

__global__ void gconv_norm(float* __restrict__ h,
                           const float* __restrict__ deg, int n_h) {
  int i = blockIdx.x * blockDim.x + threadIdx.x;
  if (i >= n_h) return;
  const int node = i >> 7;  // D_FEAT == 128
  const float dv = deg[node];
  h[i] *= 1.0f / sqrtf(fmaxf(dv, 1.0f));
}

extern "C" void kernel_launch(void* const* d_in, const int* in_sizes, int n_in,
                              void* d_out, int out_size, void* d_ws, size_t ws_size,
                              hipStream_t stream) {
  const float* feat = (const float*)d_in[0];
  const int*   src  = (const int*)d_in[1];
  const int*   dst  = (const int*)d_in[2];
  float* h   = (float*)d_out;
  float* deg = (float*)d_ws;                 // n_nodes floats of scratch

  const int n_edges = in_sizes[1];
  const int n_h     = out_size;              // n_nodes * 128
  const int n_nodes = n_h / D_FEAT;

  // 1) zero accumulator + degree (buffers are poisoned, not zeroed)
  gconv_zero<<<(n_h + 255) / 256, 256, 0, stream>>>(h, deg, n_h, n_nodes);

  // 2) edge-parallel scatter-add (1 wave per edge per iteration)
  gconv_scatter<<<(n_edges + EPB - 1) / EPB, 256, 0, stream>>>(
      feat, src, dst, h, deg, n_edges);

  // 3) h *= max(deg,1)^(-1/2)
  gconv_norm<<<(n_h + 255) / 256, 256, 0, stream>>>(h, deg, n_h);
}